// EmbZGConv_1254130450626
// MI455X (gfx1250) — compile-verified
//
#include <hip/hip_runtime.h>

#define N_NODES 100000
#define N_EDGES 1600000
#define HIDDEN  128
#define NDEG    65
#define EPS     1e-6f

typedef float v2f __attribute__((ext_vector_type(2)));
typedef float v8f __attribute__((ext_vector_type(8)));
typedef unsigned int u32x4 __attribute__((ext_vector_type(4)));
typedef int i32x4 __attribute__((ext_vector_type(4)));
typedef int i32x8 __attribute__((ext_vector_type(8)));

#if defined(__gfx1250__) && __has_builtin(__builtin_amdgcn_tensor_load_to_lds) && \
    __has_builtin(__builtin_amdgcn_s_wait_tensorcnt)
#define USE_TDM 1
#endif

// ---- workspace layout (float elements) ----
#define OFF_DEG   0                                   // 100000 f
#define OFF_HIST  100000                              // 65 u32
#define OFF_STATS 100096                              // S1,SS1,S2,SS2 = 512 f
#define OFF_S1    100608                              // N*HIDDEN = 12,800,000 f
#define ZERO_END  (OFF_S1 + N_NODES * HIDDEN)         // everything before here zeroed
#define OFF_WNORM ZERO_END                            // 1,600,000 f
#define OFF_AC    (OFF_WNORM + N_EDGES)               // a0,c0,A12,C12,A34,C34 = 768 f

__device__ __forceinline__ float atomAddF(float* p, float v) {
  return __hip_atomic_fetch_add(p, v, __ATOMIC_RELAXED, __HIP_MEMORY_SCOPE_AGENT);
}

// -------- zero ws[0..ZERO_END) and d_out --------
__global__ void k_zero(float* __restrict__ ws, float* __restrict__ out) {
  int i = blockIdx.x * blockDim.x + threadIdx.x;
  if (i < ZERO_END) ws[i] = 0.0f;
  else out[i - ZERO_END] = 0.0f;
}

// -------- degree accumulation + degree-label histogram --------
__global__ void k_deg_hist(const int* __restrict__ row, const int* __restrict__ x,
                           const float* __restrict__ ew,
                           float* __restrict__ deg, unsigned* __restrict__ hist) {
  int i = blockIdx.x * blockDim.x + threadIdx.x;
  if (i < N_EDGES) atomAddF(&deg[row[i]], ew[i]);
  if (i < N_NODES)
    __hip_atomic_fetch_add(&hist[x[i]], 1u, __ATOMIC_RELAXED, __HIP_MEMORY_SCOPE_AGENT);
}

// -------- w_norm[e] = ew[e] / clamp(deg[row[e]]) --------
__global__ void k_wnorm(const int* __restrict__ row, const float* __restrict__ ew,
                        const float* __restrict__ deg, float* __restrict__ wn) {
  int e = blockIdx.x * blockDim.x + threadIdx.x;
  if (e >= N_EDGES) return;
  float d = deg[row[e]];
  d = (d < 0.5f) ? d + 1.0f : d;
  wn[e] = ew[e] / d;
}

// -------- stats of emb[x] from histogram, fold into gn0 affine (a0,c0) --------
__global__ void k_emb_stats(const float* __restrict__ emb, const unsigned* __restrict__ hist,
                            const float* __restrict__ gw, const float* __restrict__ gb,
                            const float* __restrict__ gms,
                            float* __restrict__ a0, float* __restrict__ c0) {
  int d = threadIdx.x;
  if (d >= HIDDEN) return;
  float S = 0.f, SS = 0.f;
  for (int k = 0; k < NDEG; ++k) {
    float c = (float)hist[k];
    float e = emb[k * HIDDEN + d];
    S += c * e; SS += c * e * e;
  }
  const float inv = 1.0f / (float)N_NODES;
  float mean = S * inv, ex2 = SS * inv;
  float t = mean * gms[d];
  float var = ex2 - 2.f * t * mean + t * t;
  float stdv = sqrtf(var + EPS);
  a0[d] = gw[d] / stdv;
  c0[d] = gb[d] - gw[d] * t / stdv;
}

// -------- SpMM 1: s1[r] += w * (a0*emb[x[c]] + c0) --------
// Raw emb table (65x128 f32 = 33,280 B) DMA'd into LDS by the Tensor Data Mover;
// gn0 affine applied from per-lane registers. 512 edges per block.
__global__ void __launch_bounds__(256) k_spmm1(
    const int* __restrict__ row, const int* __restrict__ col,
    const int* __restrict__ x, const float* __restrict__ emb,
    const float* __restrict__ a0, const float* __restrict__ c0,
    const float* __restrict__ wn, float* __restrict__ s1) {
  __shared__ float embn[NDEG * HIDDEN];
  int tid = threadIdx.x;
#ifdef USE_TDM
  if (tid < 32) {   // one wave issues the TDM op (EXEC ignored; TENSORcnt per-wave)
    unsigned int ldsoff = (unsigned int)(size_t)&embn[0];        // addr[31:0] = LDS offset
    unsigned long long ga =
        (unsigned long long)(size_t)emb & ((1ull << 57) - 1ull); // 57-bit global addr
    u32x4 g0;
    g0.x = 1u;                                      // count=1, user mode, no gather
    g0.y = ldsoff;                                  // lds_addr
    g0.z = (unsigned int)ga;                        // global_addr[31:0]
    g0.w = (unsigned int)(ga >> 32) | (2u << 30);   // global_addr[56:32] | type=2
    i32x8 g1;
    g1[0] = 2 << 16;        // workgroup_mask=0, data_size=4B, no barrier/iter/pad
    g1[1] = 0x20800000;     // tensor_dim0[15:0]=8320 in bits[63:48]
    g1[2] = 1 << 16;        // tensor_dim0[31:16]=0, tensor_dim1[15:0]=1
    g1[3] = 0x20800000;     // tensor_dim1[31:16]=0, tile_dim0=8320
    g1[4] = 1;              // tile_dim1=1, tile_dim2=0
    g1[5] = 8320;           // tensor_dim0_stride[31:0]
    g1[6] = 0;              // tensor_dim0_stride[47:32], tensor_dim1_stride[15:0]
    g1[7] = 0;              // tensor_dim1_stride[47:16]
    i32x4 z4 = {0, 0, 0, 0};
    i32x8 z8 = {0, 0, 0, 0, 0, 0, 0, 0};
    __builtin_amdgcn_tensor_load_to_lds(g0, g1, z4, z4, z8, 0);
    __builtin_amdgcn_s_wait_tensorcnt(0);
  }
#else
  for (int i = tid; i < NDEG * HIDDEN; i += 256) embn[i] = emb[i];
#endif
  __syncthreads();
  int wave = tid >> 5, lane = tid & 31;
  float a0v[4], c0v[4];
#pragma unroll
  for (int j = 0; j < 4; ++j) { int d = lane + 32 * j; a0v[j] = a0[d]; c0v[j] = c0[d]; }
  for (int b = 0; b < 2; ++b) {
    int ebase = blockIdx.x * 512 + wave * 64 + b * 32;
    int e_l = ebase + lane;                 // coalesced per-lane metadata loads
    int   r_l = row[e_l];
    int   c_l = col[e_l];
    float w_l = wn[e_l];
    int   k_l = x[c_l];                     // 32 gathers issued at once
    for (int ep = 0; ep < 32; ++ep) {
      int   r = __shfl(r_l, ep, 32);
      int   k = __shfl(k_l, ep, 32);
      float w = __shfl(w_l, ep, 32);
      const float* er = &embn[k * HIDDEN];
      float* orow = &s1[r * HIDDEN];
#pragma unroll
      for (int j = 0; j < 4; ++j) {
        int d = lane + 32 * j;
        atomAddF(&orow[d], w * fmaf(a0v[j], er[d], c0v[j]));
      }
    }
  }
}

// -------- column sum / sumsq over [N_NODES, HIDDEN] via WMMA f32 16x16x4 --------
// D = ones(16x4) x B + D  => every row of D accumulates the per-column sum of 4 node rows.
__global__ void __launch_bounds__(256) k_colstats(const float* __restrict__ in,
                                                  float* __restrict__ S,
                                                  float* __restrict__ SS) {
#if __has_builtin(__builtin_amdgcn_wmma_f32_16x16x4_f32)
  int tid = threadIdx.x;
  int wave = tid >> 5, lane = tid & 31;
  int col = wave * 16 + (lane & 15);     // 8 waves -> 8 column groups of 16
  int rsel = (lane >> 4) * 2;            // lanes 16-31 take rows +2,+3
  v8f cs = {}, cq = {};
  v2f ones; ones.x = 1.0f; ones.y = 1.0f;
  for (int chunk = blockIdx.x; chunk < N_NODES / 4; chunk += gridDim.x) {
    int r = chunk * 4 + rsel;
    v2f b;
    b.x = in[r * HIDDEN + col];
    b.y = in[(r + 1) * HIDDEN + col];
    v2f b2 = b * b;
    cs = __builtin_amdgcn_wmma_f32_16x16x4_f32(false, ones, false, b,  (short)0, cs, false, false);
    cq = __builtin_amdgcn_wmma_f32_16x16x4_f32(false, ones, false, b2, (short)0, cq, false, false);
  }
  if (lane < 16) {                       // D row M=0: VGPR0, lanes 0-15, N=lane
    atomAddF(&S[col],  cs[0]);
    atomAddF(&SS[col], cq[0]);
  }
#else
  int tid = threadIdx.x;
  int col = tid & (HIDDEN - 1);
  int roff = tid >> 7;
  float s = 0.f, q = 0.f;
  for (int r = blockIdx.x * 2 + roff; r < N_NODES; r += gridDim.x * 2) {
    float v = in[r * HIDDEN + col];
    s += v; q += v * v;
  }
  atomAddF(&S[col], s);
  atomAddF(&SS[col], q);
#endif
}

// -------- chain two GraphNorms into one affine (A,C) from column stats --------
__global__ void k_affine2(const float* __restrict__ S, const float* __restrict__ SS,
                          const float* __restrict__ w1, const float* __restrict__ b1,
                          const float* __restrict__ ms1,
                          const float* __restrict__ w2, const float* __restrict__ b2,
                          const float* __restrict__ ms2,
                          float* __restrict__ A, float* __restrict__ C) {
  int d = threadIdx.x;
  if (d >= HIDDEN) return;
  const float inv = 1.0f / (float)N_NODES;
  float mean = S[d] * inv, ex2 = SS[d] * inv;
  // gn #1
  float t = mean * ms1[d];
  float var = ex2 - 2.f * t * mean + t * t;
  float stdv = sqrtf(var + EPS);
  float a1 = w1[d] / stdv;
  float c1 = b1[d] - w1[d] * t / stdv;
  // propagate stats through affine a1*h + c1
  float mean2 = a1 * mean + c1;
  float ex2b = a1 * a1 * ex2 + 2.f * a1 * c1 * mean + c1 * c1;
  // gn #2
  t = mean2 * ms2[d];
  var = ex2b - 2.f * t * mean2 + t * t;
  float std2 = sqrtf(var + EPS);
  float a2 = w2[d] / std2;
  float c2 = b2[d] - w2[d] * t / std2;
  A[d] = a2 * a1;
  C[d] = a2 * c1 + c2;
}

// -------- SpMM 2: out[r] += w * relu(A*s1[c] + C)  (gn1∘gn2∘relu fused in gather) --------
__global__ void __launch_bounds__(256) k_spmm2(
    const int* __restrict__ row, const int* __restrict__ col,
    const float* __restrict__ wn, const float* __restrict__ s1,
    const float* __restrict__ A, const float* __restrict__ C,
    float* __restrict__ out) {
  int tid = threadIdx.x;
  int wave = tid >> 5, lane = tid & 31;
  float Av[4], Cv[4];
#pragma unroll
  for (int j = 0; j < 4; ++j) { int d = lane + 32 * j; Av[j] = A[d]; Cv[j] = C[d]; }
  int ebase = blockIdx.x * 256 + wave * 32;
  int e_l = ebase + lane;                   // coalesced per-lane metadata loads
  int   r_l = row[e_l];
  int   c_l = col[e_l];
  float w_l = wn[e_l];
  for (int ep = 0; ep < 32; ++ep) {
    int   r = __shfl(r_l, ep, 32);
    int   c = __shfl(c_l, ep, 32);
    float w = __shfl(w_l, ep, 32);
    const float* srow = &s1[c * HIDDEN];
    float* orow = &out[r * HIDDEN];
#pragma unroll
    for (int j = 0; j < 4; ++j) {
      int d = lane + 32 * j;
      float v = fmaxf(fmaf(Av[j], srow[d], Cv[j]), 0.0f);
      atomAddF(&orow[d], w * v);
    }
  }
}

// -------- final elementwise: out = A34*out + C34 (gn3∘gn4) --------
__global__ void k_finish(float* __restrict__ out, const float* __restrict__ A,
                         const float* __restrict__ C) {
  int i = blockIdx.x * blockDim.x + threadIdx.x;
  int d = i & (HIDDEN - 1);
  out[i] = A[d] * out[i] + C[d];
}

extern "C" void kernel_launch(void* const* d_in, const int* in_sizes, int n_in,
                              void* d_out, int out_size, void* d_ws, size_t ws_size,
                              hipStream_t stream) {
  const int*   x    = (const int*)d_in[0];
  const int*   ei   = (const int*)d_in[1];   // [2, E]: row = ei, col = ei + E
  const float* ew   = (const float*)d_in[2];
  const float* emb  = (const float*)d_in[3];
  const float* gw   = (const float*)d_in[4]; // [5,128]
  const float* gb   = (const float*)d_in[5];
  const float* gms  = (const float*)d_in[6];
  float* out = (float*)d_out;
  float* ws  = (float*)d_ws;

  const int* row = ei;
  const int* col = ei + N_EDGES;

  float*    deg  = ws + OFF_DEG;
  unsigned* hist = (unsigned*)(ws + OFF_HIST);
  float* S1  = ws + OFF_STATS;
  float* SS1 = S1 + HIDDEN;
  float* S2  = SS1 + HIDDEN;
  float* SS2 = S2 + HIDDEN;
  float* s1  = ws + OFF_S1;
  float* wn  = ws + OFF_WNORM;
  float* a0  = ws + OFF_AC;
  float* c0  = a0 + HIDDEN;
  float* A12 = c0 + HIDDEN;
  float* C12 = A12 + HIDDEN;
  float* A34 = C12 + HIDDEN;
  float* C34 = A34 + HIDDEN;

  const int ZTOT = ZERO_END + N_NODES * HIDDEN;      // ws prefix + d_out
  k_zero<<<(ZTOT + 255) / 256, 256, 0, stream>>>(ws, out);
  k_deg_hist<<<N_EDGES / 256, 256, 0, stream>>>(row, x, ew, deg, hist);
  k_emb_stats<<<1, 128, 0, stream>>>(emb, hist, gw, gb, gms, a0, c0);
  k_wnorm<<<N_EDGES / 256, 256, 0, stream>>>(row, ew, deg, wn);
  k_spmm1<<<N_EDGES / 512, 256, 0, stream>>>(row, col, x, emb, a0, c0, wn, s1);
  k_colstats<<<256, 256, 0, stream>>>(s1, S1, SS1);
  k_affine2<<<1, 128, 0, stream>>>(S1, SS1,
                                   gw + 1 * HIDDEN, gb + 1 * HIDDEN, gms + 1 * HIDDEN,
                                   gw + 2 * HIDDEN, gb + 2 * HIDDEN, gms + 2 * HIDDEN,
                                   A12, C12);
  k_spmm2<<<N_EDGES / 256, 256, 0, stream>>>(row, col, wn, s1, A12, C12, out);
  k_colstats<<<256, 256, 0, stream>>>(out, S2, SS2);
  k_affine2<<<1, 128, 0, stream>>>(S2, SS2,
                                   gw + 3 * HIDDEN, gb + 3 * HIDDEN, gms + 3 * HIDDEN,
                                   gw + 4 * HIDDEN, gb + 4 * HIDDEN, gms + 4 * HIDDEN,
                                   A34, C34);
  k_finish<<<(N_NODES * HIDDEN) / 256, 256, 0, stream>>>(out, A34, C34);
}